// CBFGraphNet_53678501265901
// MI455X (gfx1250) — compile-verified
//
#include <hip/hip_runtime.h>

// ---------------------------------------------------------------------------
// CBF GraphNet, dead-code-eliminated to node 0's dependence cone.
//   agg16 = sum_{e: recv[e]==0} edge_feats[e]   (16 floats),  deg = count
//   s64   = agg16 @ edge_W + deg*edge_b
//   n     = node_feats[0] @ node_W + node_b
//   for l in {0,1}:
//     x = n + (s64 @ mW_l + deg*mb_l)
//     n = relu(x @ uW_l + ub_l)
//   h   = relu(n @ cbf_W1 + cbf_b1)
//   out = h @ cbf_W2 + cbf_b2
// Dense head runs on one wave32 with V_WMMA_F32_16X16X4_F32 chains.
// A-operand carries the vector broadcast to all 16 rows (no predication,
// EXEC stays all-ones); row 0 of D is the result.
// ---------------------------------------------------------------------------

typedef __attribute__((ext_vector_type(2))) float v2f;
typedef __attribute__((ext_vector_type(8))) float v8f;

#define RED_TPB   256
#define RED_EPT   4          // edges per thread per stride step (int4 load)
#define RED_BLKS  512        // partial buffer = 512*17*4 = 34.8 KB in d_ws

// ---------------------------------------------------------------------------
// Kernel 1: deterministic masked reduction over edges with receiver == 0.
// Streams `receivers` with int4 loads; gathers the rare matching edge_feats
// rows; block-local LDS reduction in fixed order (no atomics).
// partial[b*17 + 0..15] = block b's feature sums, partial[b*17+16] = count.
// ---------------------------------------------------------------------------
__global__ __launch_bounds__(RED_TPB)
void reduce_r0(const int* __restrict__ recv,
               const float* __restrict__ ef,
               float* __restrict__ partial,
               int E) {
  float acc[16];
#pragma unroll
  for (int j = 0; j < 16; ++j) acc[j] = 0.0f;
  float cnt = 0.0f;

  const long long stride = (long long)RED_BLKS * RED_TPB * RED_EPT;
  for (long long base = ((long long)blockIdx.x * RED_TPB + threadIdx.x) * RED_EPT;
       base < E; base += stride) {
    // prefetch next stride chunk of the receiver stream (global_prefetch_b8)
    if (base + stride < E) __builtin_prefetch(recv + base + stride, 0, 1);

    if (base + RED_EPT <= E) {
      const int4 r4 = *(const int4*)(recv + base);
      const int rr[RED_EPT] = { r4.x, r4.y, r4.z, r4.w };
#pragma unroll
      for (int q = 0; q < RED_EPT; ++q) {
        if (rr[q] == 0) {
          const float4* p = (const float4*)(ef + (size_t)(base + q) * 16);
          const float4 f0 = p[0], f1 = p[1], f2 = p[2], f3 = p[3];
          acc[ 0] += f0.x; acc[ 1] += f0.y; acc[ 2] += f0.z; acc[ 3] += f0.w;
          acc[ 4] += f1.x; acc[ 5] += f1.y; acc[ 6] += f1.z; acc[ 7] += f1.w;
          acc[ 8] += f2.x; acc[ 9] += f2.y; acc[10] += f2.z; acc[11] += f2.w;
          acc[12] += f3.x; acc[13] += f3.y; acc[14] += f3.z; acc[15] += f3.w;
          cnt += 1.0f;
        }
      }
    } else {
      for (long long q = base; q < E; ++q) {
        if (recv[q] == 0) {
          const float* p = ef + (size_t)q * 16;
#pragma unroll
          for (int j = 0; j < 16; ++j) acc[j] += p[j];
          cnt += 1.0f;
        }
      }
    }
  }

  // deterministic block reduction: column j summed in fixed thread order
  __shared__ float red[RED_TPB * 17];
  const int tid = threadIdx.x;
#pragma unroll
  for (int j = 0; j < 16; ++j) red[tid * 17 + j] = acc[j];
  red[tid * 17 + 16] = cnt;
  __syncthreads();
  if (tid < 17) {
    float s = 0.0f;
    for (int t = 0; t < RED_TPB; ++t) s += red[t * 17 + tid];
    partial[(size_t)blockIdx.x * 17 + tid] = s;
  }
}

// ---------------------------------------------------------------------------
// fp32 y[N] = x[K] @ W[K,N] via chained V_WMMA_F32_16X16X4_F32.
//  * A operand: x broadcast into all 16 rows (unconditional ds_load_b64 of
//    the lane-half's K pair) -> every row of D equals y; we read row 0.
//  * B operand: W^T staged in LDS (coalesced float4 global loads), so
//    (W[k0][n], W[k0+1][n]) is one aligned ds_load_b64.
// ---------------------------------------------------------------------------
template<int K, int N>
__device__ inline void matvec_wmma(const float* __restrict__ W,
                                   const float* __restrict__ s_x,
                                   float* __restrict__ s_y,
                                   float* __restrict__ s_wt) {
  const int lane = threadIdx.x & 31;
  __syncthreads();                       // prior users of s_wt are done
  // stage W^T: s_wt[n*K + k] = W[k*N + n]
  for (int idx = lane * 4; idx < K * N; idx += 32 * 4) {
    const float4 w4 = *(const float4*)(W + idx);
    const int k = idx / N, n = idx % N;  // N | 4, so the quad shares k
    s_wt[(n + 0) * K + k] = w4.x;
    s_wt[(n + 1) * K + k] = w4.y;
    s_wt[(n + 2) * K + k] = w4.z;
    s_wt[(n + 3) * K + k] = w4.w;
  }
  __syncthreads();

  const int row  = lane & 15;
  const int koff = (lane >> 4) << 1;     // lanes 0-15: K={0,1}; 16-31: K={2,3}
#pragma unroll
  for (int c = 0; c < N / 16; ++c) {
    v8f acc = {};
#pragma unroll
    for (int s = 0; s < K / 4; ++s) {
      const int k0 = 4 * s + koff;
      const v2f a = *(const v2f*)(s_x + k0);                        // broadcast
      const v2f b = *(const v2f*)(s_wt + (c * 16 + row) * K + k0);  // W^T pair
      acc = __builtin_amdgcn_wmma_f32_16x16x4_f32(
          /*neg_a=*/false, a, /*neg_b=*/false, b,
          /*c_mod=*/(short)0, acc, /*reuse_a=*/false, /*reuse_b=*/false);
    }
    if (lane < 16) s_y[c * 16 + lane] = acc[0];                     // row 0 of D
  }
}

// ---------------------------------------------------------------------------
// Kernel 2: single wave32. Sums partials in fixed order, then runs the
// whole dense head with fp32 WMMA.
// ---------------------------------------------------------------------------
__global__ __launch_bounds__(32)
void cbf_head(const float* __restrict__ node_feats,
              const float* __restrict__ node_W, const float* __restrict__ node_b,
              const float* __restrict__ edge_W, const float* __restrict__ edge_b,
              const float* __restrict__ mW0, const float* __restrict__ mb0,
              const float* __restrict__ mW1, const float* __restrict__ mb1,
              const float* __restrict__ uW0, const float* __restrict__ ub0,
              const float* __restrict__ uW1, const float* __restrict__ ub1,
              const float* __restrict__ cW1, const float* __restrict__ cb1,
              const float* __restrict__ cW2, const float* __restrict__ cb2,
              const float* __restrict__ partial,
              float* __restrict__ out) {
  __shared__ alignas(16) float s_wt[64 * 64];   // W^T staging (16 KB)
  __shared__ alignas(16) float s_x[64], s_n[64], s_s[64], s_t[64];
  __shared__ alignas(16) float s_a[17];
  const int lane = threadIdx.x;

  // 1) combine per-block partials (fixed order -> deterministic);
  //    unrolled for memory-level parallelism
  if (lane < 17) {
    float s = 0.0f;
#pragma unroll 8
    for (int b = 0; b < RED_BLKS; ++b) s += partial[(size_t)b * 17 + lane];
    s_a[lane] = s;
  }
  __syncthreads();
  const float deg = s_a[16];

  // 2) n = node_feats[0] @ node_W + node_b
  if (lane < 16) s_x[lane] = node_feats[lane];
  matvec_wmma<16, 64>(node_W, s_x, s_n, s_wt);   // syncs internally
  __syncthreads();
  for (int i = lane; i < 64; i += 32) s_n[i] += node_b[i];

  // 3) s64 = agg16 @ edge_W + deg*edge_b   (aggregated encoded-edge sum)
  if (lane < 16) s_x[lane] = s_a[lane];
  matvec_wmma<16, 64>(edge_W, s_x, s_s, s_wt);
  __syncthreads();
  for (int i = lane; i < 64; i += 32) s_s[i] += deg * edge_b[i];

  // 4) two message-passing layers (node-0 cone only)
  const float* mW[2] = { mW0, mW1 };
  const float* mb[2] = { mb0, mb1 };
  const float* uW[2] = { uW0, uW1 };
  const float* ub[2] = { ub0, ub1 };
#pragma unroll
  for (int l = 0; l < 2; ++l) {
    matvec_wmma<64, 64>(mW[l], s_s, s_t, s_wt);      // s64 @ mW_l
    __syncthreads();
    for (int i = lane; i < 64; i += 32)
      s_x[i] = s_n[i] + s_t[i] + deg * mb[l][i];     // n + aggregated_l
    matvec_wmma<64, 64>(uW[l], s_x, s_t, s_wt);      // x @ uW_l
    __syncthreads();
    for (int i = lane; i < 64; i += 32)
      s_n[i] = fmaxf(s_t[i] + ub[l][i], 0.0f);       // relu(. + ub_l)
  }

  // 5) CBF head: h = relu(n @ cbf_W1 + cbf_b1); out = h @ cbf_W2 + cbf_b2
  matvec_wmma<64, 32>(cW1, s_n, s_t, s_wt);
  __syncthreads();
  const float h = fmaxf(s_t[lane] + cb1[lane], 0.0f);  // 32 outputs, 32 lanes
  float p = h * cW2[lane];
#pragma unroll
  for (int off = 16; off > 0; off >>= 1) p += __shfl_down(p, off, 32);
  if (lane == 0) out[0] = p + cb2[0];
}

// ---------------------------------------------------------------------------
extern "C" void kernel_launch(void* const* d_in, const int* in_sizes, int n_in,
                              void* d_out, int out_size, void* d_ws, size_t ws_size,
                              hipStream_t stream) {
  const float* node_feats = (const float*)d_in[0];
  const float* edge_feats = (const float*)d_in[1];
  const int*   receivers  = (const int*)  d_in[2];
  const float* node_W = (const float*)d_in[3];
  const float* node_b = (const float*)d_in[4];
  const float* edge_W = (const float*)d_in[5];
  const float* edge_b = (const float*)d_in[6];
  const float* msg_W0 = (const float*)d_in[7];
  const float* msg_b0 = (const float*)d_in[8];
  const float* msg_W1 = (const float*)d_in[9];
  const float* msg_b1 = (const float*)d_in[10];
  const float* upd_W0 = (const float*)d_in[11];
  const float* upd_b0 = (const float*)d_in[12];
  const float* upd_W1 = (const float*)d_in[13];
  const float* upd_b1 = (const float*)d_in[14];
  const float* cbf_W1 = (const float*)d_in[15];
  const float* cbf_b1 = (const float*)d_in[16];
  const float* cbf_W2 = (const float*)d_in[17];
  const float* cbf_b2 = (const float*)d_in[18];

  const int E = in_sizes[2];
  float* partial = (float*)d_ws;     // RED_BLKS * 17 floats

  reduce_r0<<<RED_BLKS, RED_TPB, 0, stream>>>(receivers, edge_feats, partial, E);
  cbf_head<<<1, 32, 0, stream>>>(node_feats, node_W, node_b, edge_W, edge_b,
                                 msg_W0, msg_b0, msg_W1, msg_b1,
                                 upd_W0, upd_b0, upd_W1, upd_b1,
                                 cbf_W1, cbf_b1, cbf_W2, cbf_b2,
                                 partial, (float*)d_out);
}